// SACNN_84464826843619
// MI455X (gfx1250) — compile-verified
//
#include <hip/hip_runtime.h>
#include <hip/hip_bf16.h>
#include <math.h>

typedef __attribute__((ext_vector_type(2))) float v2f;
typedef __attribute__((ext_vector_type(4))) float v4f;
typedef __attribute__((ext_vector_type(8))) float v8f;

#define KNN 25
#define NB  16
#define NP  2048

// ---------------------------------------------------------------------------
// f32 WMMA: D = A(16x4) * B(4x16) + C. Fragment layout per CDNA5 ISA 7.12.2:
//   lanes 0-15 hold K = {k0,k0+1}, lanes 16-31 hold K = {k0+2,k0+3}
//   C/D: VGPR r <-> rows r (lanes 0-15) / r+8 (lanes 16-31), col = lane&15
// ---------------------------------------------------------------------------
__device__ __forceinline__ v8f wmma4(v2f a, v2f b, v8f c) {
  return __builtin_amdgcn_wmma_f32_16x16x4_f32(
      false, a, false, b, (short)0, c, false, false);
}

// Software-pipelined K-loop, one A fragment feeding two N-tiles (two
// independent WMMA chains: overlaps the D->A/B hazard window and halves A
// traffic). klen must be a multiple of 4 and >= 8. Works for both global
// and LDS operands (forceinline preserves address-space inference).
__device__ __forceinline__ void seg2(const float* As, const float* W0,
                                     const float* W1, int klen, int hl,
                                     v8f& acc0, v8f& acc1) {
  int kk = 2 * hl;
  v2f a  = *(const v2f*)(As + kk);
  v2f b0 = *(const v2f*)(W0 + kk);
  v2f b1 = *(const v2f*)(W1 + kk);
  for (int k = 4; k < klen; k += 4) {
    int kn = k + 2 * hl;
    v2f na  = *(const v2f*)(As + kn);
    v2f nb0 = *(const v2f*)(W0 + kn);
    v2f nb1 = *(const v2f*)(W1 + kn);
    acc0 = wmma4(a, b0, acc0);
    acc1 = wmma4(a, b1, acc1);
    a = na; b0 = nb0; b1 = nb1;
  }
  acc0 = wmma4(a, b0, acc0);
  acc1 = wmma4(a, b1, acc1);
}

__device__ __forceinline__ void seg1(const float* As, const float* W0,
                                     int klen, int hl, v8f& acc0) {
  int kk = 2 * hl;
  v2f a  = *(const v2f*)(As + kk);
  v2f b0 = *(const v2f*)(W0 + kk);
  for (int k = 4; k < klen; k += 4) {
    int kn = k + 2 * hl;
    v2f na  = *(const v2f*)(As + kn);
    v2f nb0 = *(const v2f*)(W0 + kn);
    acc0 = wmma4(a, b0, acc0);
    a = na; b0 = nb0;
  }
  acc0 = wmma4(a, b0, acc0);
}

// ---------------------------------------------------------------------------
// KNN: one wave32 per point. Rank key = |x_m|^2 - 2 x_n.x_m (order-preserving
// shift of the true distance). float4-vectorized distance loop (C % 4 == 0
// layers); per-lane top-25 in registers, then 25-round shfl-butterfly merge.
// Feature tensor (<=32MB) is L2-resident (192MB L2), so the O(N^2 C)
// re-reads stream from L2, not HBM.
// ---------------------------------------------------------------------------
__global__ __launch_bounds__(256) void knn_kernel(const float* __restrict__ F,
                                                  int C,
                                                  int* __restrict__ idx_out) {
  __shared__ float xn[8][128];
  const int wave = threadIdx.x >> 5;
  const int lane = threadIdx.x & 31;
  const int row  = blockIdx.x * 8 + wave;   // b*NP + n
  const int b    = row / NP;

  const float* xr = F + (size_t)row * C;
  for (int c = lane; c < C; c += 32) xn[wave][c] = xr[c];
  // wave-private LDS region; DS ops are in-order within a wave.

  float kd[KNN]; int ki[KNN];
#pragma unroll
  for (int t = 0; t < KNN; t++) { kd[t] = INFINITY; ki[t] = -1; }
  float wmax = INFINITY; int wpos = 0;

  const float* Fb = F + (size_t)b * NP * C;
  const bool vec4 = (C & 3) == 0;
  for (int m = lane; m < NP; m += 32) {
    const float* xm = Fb + (size_t)m * C;
    float dot = 0.f, s2 = 0.f;
    if (vec4) {
      for (int c = 0; c < C; c += 4) {
        v4f v = *(const v4f*)(xm + c);
        v4f u = *(const v4f*)(&xn[wave][c]);
#pragma unroll
        for (int j = 0; j < 4; j++) {
          dot = fmaf(u[j], v[j], dot);
          s2  = fmaf(v[j], v[j], s2);
        }
      }
    } else {
      for (int c = 0; c < C; c++) {
        float v = xm[c];
        dot = fmaf(xn[wave][c], v, dot);
        s2  = fmaf(v, v, s2);
      }
    }
    float key = s2 - 2.f * dot;
    if (key < wmax) {
      kd[wpos] = key; ki[wpos] = m;
      wmax = kd[0]; wpos = 0;
#pragma unroll
      for (int t = 1; t < KNN; t++)
        if (kd[t] > wmax) { wmax = kd[t]; wpos = t; }
    }
  }

  int* orow = idx_out + (size_t)row * KNN;
  for (int j = 0; j < KNN; j++) {
    float bv = kd[0]; int bp = 0;
#pragma unroll
    for (int t = 1; t < KNN; t++)
      if (kd[t] < bv) { bv = kd[t]; bp = t; }
    int bi = ki[bp];
    float gv = bv; int gi = bi;
    for (int off = 16; off; off >>= 1) {
      float ov = __shfl_xor(gv, off, 32);
      int   oi = __shfl_xor(gi, off, 32);
      if (ov < gv || (ov == gv && oi < gi)) { gv = ov; gi = oi; }
    }
    if (bv == gv && bi == gi) kd[bp] = INFINITY;  // pop winner's entry
    if (lane == 0) orow[j] = gi;
  }
}

// ---------------------------------------------------------------------------
// X = feats0[:, :, None, :] - gather(feats0, idx0)   (layer-0 only, reused)
// ---------------------------------------------------------------------------
__global__ __launch_bounds__(256) void xdiff_kernel(const float* __restrict__ F0,
                                                    const int* __restrict__ idx,
                                                    float* __restrict__ X) {
  int t = blockIdx.x * 256 + threadIdx.x;       // over B*N*K
  const int total = NB * NP * KNN;
  if (t >= total) return;
  int bn = t / KNN;
  int b  = bn / NP;
  int m  = idx[t];
  const float* pn = F0 + (size_t)bn * 3;
  const float* pm = F0 + ((size_t)b * NP + m) * 3;
  float* xo = X + (size_t)t * 3;
  xo[0] = pn[0] - pm[0];
  xo[1] = pn[1] - pm[1];
  xo[2] = pn[2] - pm[2];
}

// ---------------------------------------------------------------------------
// Fused edge-conv per point. ef (25 rows zero-padded to 32, K zero-padded to
// CkP) and edge_w (zero-padded to CkP) staged in LDS -> branch-free pipelined
// WMMA loops (uniform tile-count cases hoisted OUT of the k-loop).
// Max/mean over the 25 valid rows reduced straight out of the accumulators.
// LDS ~85KB/WG (WGP has 320KB).
// ---------------------------------------------------------------------------
__global__ __launch_bounds__(256) void edge_kernel(const float* __restrict__ F,
                            const int* __restrict__ idx,
                            const float* __restrict__ X,
                            const float* __restrict__ aug_w,
                            const float* __restrict__ aug_b,
                            const float* __restrict__ edge_w,
                            const float* __restrict__ edge_b,
                            int cin, int half,
                            float* __restrict__ agg) {
  __shared__ float ef[32 * 132];    // A tile, max (cin+3) padded = 132
  __shared__ float ew[128 * 132];   // B tile (edge_w), zero-padded K
  __shared__ float pm[2 * 128];     // per-Mtile partial max
  __shared__ float ps[2 * 128];     // per-Mtile partial sum
  __shared__ float tri[3];

  const int bn   = blockIdx.x;      // b*NP + n
  const int b    = bn / NP;
  const int Cact = cin + 3;
  const int CkP  = (Cact + 3) & ~3;
  const int tid  = threadIdx.x;

  if (tid < 3) {
    float s = aug_b[tid];
    const float* xr = F + (size_t)bn * cin;
    for (int c = 0; c < cin; c++) s = fmaf(aug_w[tid * cin + c], xr[c], s);
    tri[tid] = s;
  }
  // stage edge_w with zero padding: keeps the WMMA loop guard-free
  for (int e = tid; e < half * CkP; e += 256) {
    int col = e / CkP, j = e % CkP;
    ew[e] = (j < Cact) ? edge_w[(size_t)col * Cact + j] : 0.f;
  }
  __syncthreads();   // tri ready

  const int* irow = idx + (size_t)bn * KNN;
  for (int e = tid; e < 32 * CkP; e += 256) {
    int r = e / CkP, j = e % CkP;
    float v = 0.f;
    if (r < KNN) {
      if (j < 3)
        v = X[((size_t)bn * KNN + r) * 3 + j] + tri[j];
      else if (j < Cact)
        v = F[((size_t)b * NP + irow[r]) * cin + (j - 3)];
    }
    ef[e] = v;
  }
  __syncthreads();

  const int wave = tid >> 5, lane = tid & 31;
  const int hl = lane >> 4, l = lane & 15;
  const int ntN = half >> 4;          // 2 / 4 / 8
  const int mt  = wave >> 2;          // 0..1
  const int nt0 = wave & 3;
  const int nt1 = nt0 + 4;
  const bool h0 = nt0 < ntN;
  const bool h1 = nt1 < ntN;          // wave-uniform; h1 implies h0

  v8f acc0 = {}, acc1 = {};
  const float* Ar = &ef[(mt * 16 + l) * CkP];
  const float* B0 = &ew[(nt0 * 16 + l) * CkP];
  const float* B1 = &ew[(nt1 * 16 + l) * CkP];
  if (h1)      seg2(Ar, B0, B1, CkP, hl, acc0, acc1);
  else if (h0) seg1(Ar, B0, CkP, hl, acc0);

  // reduce rows of each 16x16 tile: rows rr = mt*16 + r + 8*hl, valid if <25
  auto tile_reduce = [&](v8f acc, int nt) {
    float mx = -INFINITY, sm = 0.f;
#pragma unroll
    for (int r = 0; r < 8; r++) {
      int rr = mt * 16 + r + 8 * hl;
      float v = acc[r];
      if (rr < KNN) { mx = fmaxf(mx, v); sm += v; }
    }
    mx = fmaxf(mx, __shfl_xor(mx, 16, 32));
    sm += __shfl_xor(sm, 16, 32);
    if (lane < 16) {
      int col = nt * 16 + l;
      pm[mt * 128 + col] = mx;
      ps[mt * 128 + col] = sm;
    }
  };
  if (h0) tile_reduce(acc0, nt0);
  if (h1) tile_reduce(acc1, nt1);
  __syncthreads();

  const float inv = 1.f / (float)KNN;
  for (int c = tid; c < half; c += 256) {
    float bias = edge_b[c];
    float mx = fmaxf(pm[c], pm[128 + c]);
    float sm = ps[c] + ps[128 + c];
    agg[(size_t)bn * (2 * half) + c]        = mx + bias;
    agg[(size_t)bn * (2 * half) + half + c] = sm * inv + bias;
  }
}

// ---------------------------------------------------------------------------
// Generic GEMM: C = A(MxK) @ W(NxK)^T + bias [+C if accum] [relu].
// ALIGNED (M,N mult-16, N/16 even, K mult-4 >= 8): one wave computes TWO
// 16x16 N-tiles sharing one A fragment, pipelined b64 loads, operands stream
// from the 192MB L2. Fallback: guarded scalar loop, one tile per wave.
// ---------------------------------------------------------------------------
template <bool ALIGNED>
__global__ __launch_bounds__(256) void gemm_kernel(const float* __restrict__ A,
                            const float* __restrict__ W,
                            const float* __restrict__ bias,
                            float* __restrict__ C,
                            int M, int N, int Kd, int accum, int relu) {
  const int wid  = (blockIdx.x * blockDim.x + threadIdx.x) >> 5;
  const int lane = threadIdx.x & 31;
  const int hl = lane >> 4, l = lane & 15;
  const int ntN = (N + 15) >> 4;
  const int ntM = (M + 15) >> 4;

  if (ALIGNED) {
    const int ntH = ntN >> 1;
    if (wid >= ntM * ntH) return;   // wave-uniform: EXEC all-ones for WMMA
    const int mt = wid / ntH, ntp = wid % ntH;
    const int row  = mt * 16 + l;
    const int col0 = ntp * 16 + l;
    const int col1 = (ntp + ntH) * 16 + l;
    const float* Ap = A + (size_t)row * Kd;
    const float* W0 = W + (size_t)col0 * Kd;
    const float* W1 = W + (size_t)col1 * Kd;
    __builtin_prefetch(Ap, 0, 3);           // global_prefetch_b8
    __builtin_prefetch(W0, 0, 3);
    __builtin_prefetch(W1, 0, 3);
    v8f acc0 = {}, acc1 = {};
    seg2(Ap, W0, W1, Kd, hl, acc0, acc1);
#pragma unroll
    for (int r = 0; r < 8; r++) {
      int rr = mt * 16 + r + 8 * hl;
      float v0 = acc0[r] + bias[col0];
      float v1 = acc1[r] + bias[col1];
      if (accum) { v0 += C[(size_t)rr * N + col0]; v1 += C[(size_t)rr * N + col1]; }
      if (relu)  { v0 = fmaxf(v0, 0.f); v1 = fmaxf(v1, 0.f); }
      C[(size_t)rr * N + col0] = v0;
      C[(size_t)rr * N + col1] = v1;
    }
  } else {
    if (wid >= ntM * ntN) return;
    const int mt = wid / ntN, nt = wid % ntN;
    const int row = mt * 16 + l;
    const int col = nt * 16 + l;
    v8f acc = {};
    for (int k0 = 0; k0 < Kd; k0 += 4) {
      int kk = k0 + 2 * hl;
      bool rok = row < M, cok = col < N;
      v2f a, bb;
      a[0]  = (rok && kk     < Kd) ? A[(size_t)row * Kd + kk]     : 0.f;
      a[1]  = (rok && kk + 1 < Kd) ? A[(size_t)row * Kd + kk + 1] : 0.f;
      bb[0] = (cok && kk     < Kd) ? W[(size_t)col * Kd + kk]     : 0.f;
      bb[1] = (cok && kk + 1 < Kd) ? W[(size_t)col * Kd + kk + 1] : 0.f;
      acc = wmma4(a, bb, acc);
    }
#pragma unroll
    for (int r = 0; r < 8; r++) {
      int rr = mt * 16 + r + 8 * hl;
      if (rr < M && col < N) {
        float v = acc[r] + bias[col];
        if (accum) v += C[(size_t)rr * N + col];
        if (relu)  v = fmaxf(v, 0.f);
        C[(size_t)rr * N + col] = v;
      }
    }
  }
}

// ---------------------------------------------------------------------------
// MLP (concat(f1,f2,f3)=448 -> 1024): two N-tiles per wave, the concat handled
// as three clean per-segment loops (64/128/256, all mult-8), fused
// deterministic partial pooling in the epilogue.
// ---------------------------------------------------------------------------
__global__ __launch_bounds__(256) void mlp_pool_kernel(
    const float* __restrict__ f1, const float* __restrict__ f2,
    const float* __restrict__ f3, const float* __restrict__ Wm,
    const float* __restrict__ bm, float* __restrict__ pmax,
    float* __restrict__ psum) {
  const int Kd = 448;
  const int ntH = 32;                              // 1024/16/2
  const int wid  = (blockIdx.x * blockDim.x + threadIdx.x) >> 5;
  const int lane = threadIdx.x & 31;
  const int hl = lane >> 4, l = lane & 15;
  const int tiles = (NB * NP / 16) * ntH;
  if (wid >= tiles) return;
  const int mt = wid / ntH, ntp = wid % ntH;
  const int row  = mt * 16 + l;
  const int col0 = ntp * 16 + l;
  const int col1 = (ntp + ntH) * 16 + l;
  const float* W0 = Wm + (size_t)col0 * Kd;
  const float* W1 = Wm + (size_t)col1 * Kd;
  __builtin_prefetch(W0, 0, 3);
  __builtin_prefetch(W1, 0, 3);

  v8f acc0 = {}, acc1 = {};
  seg2(f1 + (size_t)row * 64,  W0,       W1,       64,  hl, acc0, acc1);
  seg2(f2 + (size_t)row * 128, W0 + 64,  W1 + 64,  128, hl, acc0, acc1);
  seg2(f3 + (size_t)row * 256, W0 + 192, W1 + 192, 256, hl, acc0, acc1);

  auto emit = [&](v8f acc, int col) {
    const float bias = bm[col];
    float mx = -INFINITY, sm = 0.f;
#pragma unroll
    for (int r = 0; r < 8; r++) {
      float v = acc[r] + bias;
      mx = fmaxf(mx, v);
      sm += v;
    }
    mx = fmaxf(mx, __shfl_xor(mx, 16, 32));
    sm += __shfl_xor(sm, 16, 32);
    if (lane < 16) {
      pmax[(size_t)mt * 1024 + col] = mx;
      psum[(size_t)mt * 1024 + col] = sm;
    }
  };
  emit(acc0, col0);
  emit(acc1, col1);
}

__global__ __launch_bounds__(256) void pool_reduce_kernel(
    const float* __restrict__ pmax, const float* __restrict__ psum,
    float* __restrict__ pooled) {
  int t = blockIdx.x * 256 + threadIdx.x;
  if (t >= NB * 1024) return;
  int b = t / 1024, c = t % 1024;
  float mx = -INFINITY, sm = 0.f;
  const int tilesPerB = NP / 16;                    // 128
  for (int mt = b * tilesPerB; mt < (b + 1) * tilesPerB; mt++) {
    mx = fmaxf(mx, pmax[(size_t)mt * 1024 + c]);
    sm += psum[(size_t)mt * 1024 + c];
  }
  pooled[(size_t)b * 2048 + c]        = mx;
  pooled[(size_t)b * 2048 + 1024 + c] = sm * (1.f / (float)NP);
}

// ---------------------------------------------------------------------------
// Host orchestration. Input order: features, then sa0..sa2 (aug_w,aug_b,
// edge_w,edge_b,last_w,last_b,proj_w,proj_b), mlp_w, mlp_b, cls0..3 (w,b).
// ---------------------------------------------------------------------------
extern "C" void kernel_launch(void* const* d_in, const int* in_sizes, int n_in,
                              void* d_out, int out_size, void* d_ws, size_t ws_size,
                              hipStream_t stream) {
  (void)in_sizes; (void)n_in; (void)out_size; (void)ws_size;
  const float* feats = (const float*)d_in[0];
  auto P = [&](int i) { return (const float*)d_in[i]; };

  char* w = (char*)d_ws;
  size_t off = 0;
  auto carve = [&](size_t bytes) -> void* {
    void* p = w + off;
    off = (off + bytes + 255) & ~(size_t)255;
    return p;
  };
  int*   idxb   = (int*)  carve((size_t)NB * NP * KNN * 4);
  float* X      = (float*)carve((size_t)NB * NP * KNN * 3 * 4);
  float* f1     = (float*)carve((size_t)NB * NP * 64 * 4);
  float* f2     = (float*)carve((size_t)NB * NP * 128 * 4);
  float* f3     = (float*)carve((size_t)NB * NP * 256 * 4);
  float* agg    = (float*)carve((size_t)NB * NP * 256 * 4);
  float* pmax   = (float*)carve((size_t)(NB * NP / 16) * 1024 * 4);
  float* psum   = (float*)carve((size_t)(NB * NP / 16) * 1024 * 4);
  float* pooled = (float*)carve((size_t)NB * 2048 * 4);
  float* c0     = (float*)carve((size_t)NB * 512 * 4);
  float* c1     = (float*)carve((size_t)NB * 256 * 4);
  float* c2     = (float*)carve((size_t)NB * 128 * 4);

  // aligned: M,N mult-16, N/16 even, K mult-4 (>=8) -> 2 N-tiles per wave
  auto gemm_a = [&](const float* A, const float* Wt, const float* B, float* C,
                    int M, int N, int K, int acc, int relu) {
    int tiles = (M / 16) * (N / 32);
    gemm_kernel<true><<<(tiles + 7) / 8, 256, 0, stream>>>(A, Wt, B, C, M, N, K,
                                                           acc, relu);
  };
  auto gemm_u = [&](const float* A, const float* Wt, const float* B, float* C,
                    int M, int N, int K, int acc, int relu) {
    int tiles = ((M + 15) / 16) * ((N + 15) / 16);
    gemm_kernel<false><<<(tiles + 7) / 8, 256, 0, stream>>>(A, Wt, B, C, M, N, K,
                                                            acc, relu);
  };

  const int cins[3]  = {3, 64, 128};
  const int couts[3] = {64, 128, 256};
  const float* fin = feats;
  float* fouts[3] = {f1, f2, f3};

  for (int i = 0; i < 3; i++) {
    int base = 1 + i * 8;
    int cin = cins[i], cout = couts[i], half = cout / 2;

    knn_kernel<<<NB * NP / 8, 256, 0, stream>>>(fin, cin, idxb);
    if (i == 0) {
      int total = NB * NP * KNN;
      xdiff_kernel<<<(total + 255) / 256, 256, 0, stream>>>(feats, idxb, X);
    }
    edge_kernel<<<NB * NP, 256, 0, stream>>>(fin, idxb, X,
                                             P(base + 0), P(base + 1),
                                             P(base + 2), P(base + 3),
                                             cin, half, agg);
    gemm_a(agg, P(base + 4), P(base + 5), fouts[i], NB * NP, cout, cout, 0, 0);
    if (i == 0)
      gemm_u(fin, P(base + 6), P(base + 7), fouts[i], NB * NP, cout, cin, 1, 0);
    else
      gemm_a(fin, P(base + 6), P(base + 7), fouts[i], NB * NP, cout, cin, 1, 0);
    fin = fouts[i];
  }

  {
    int tiles = (NB * NP / 16) * 32;
    mlp_pool_kernel<<<(tiles + 7) / 8, 256, 0, stream>>>(f1, f2, f3, P(25), P(26),
                                                         pmax, psum);
  }
  pool_reduce_kernel<<<(NB * 1024 + 255) / 256, 256, 0, stream>>>(pmax, psum, pooled);

  gemm_a(pooled, P(27), P(28), c0, NB, 512, 2048, 0, 1);
  gemm_a(c0,     P(29), P(30), c1, NB, 256, 512,  0, 1);
  gemm_a(c1,     P(31), P(32), c2, NB, 128, 256,  0, 1);
  gemm_u(c2,     P(33), P(34), (float*)d_out, NB, 40, 128, 0, 0);
}